// BarycenterNorm_2791728742478
// MI455X (gfx1250) — compile-verified
//
#include <hip/hip_runtime.h>

// ---------------------------------------------------------------------------
// BarycenterNorm (eval-mode) for MI455X / gfx1250, wave32, WMMA f32 16x16x4.
// Replaces per-sample eigh with matmul-only iterations so all heavy math runs
// on the matrix pipes:
//   logm : inverse scaling-and-squaring (3x Newton-Schulz sqrt + log series)
//   expm : scaling-and-squaring (Taylor-Horner + repeated squaring)
// One workgroup (8 wave32) per 64x64 sample, fully LDS-resident.
// ---------------------------------------------------------------------------

typedef __attribute__((ext_vector_type(2))) float v2f;
typedef __attribute__((ext_vector_type(8))) float v8f;

#define NTHREADS 256
#define SLDS 76                 // padded LDS row stride (floats):
                                // bank-conflict-free for all WMMA frag patterns
#define MAT_LDS (64 * SLDS)
#define IDX(e) ((((e) >> 6) * SLDS) + ((e) & 63))

// ---- 64x64 GEMM, operands/result in LDS. C = A @ B (or A @ B^T). ----------
// 16 output tiles of 16x16; each of the 8 waves owns 2 tiles; K swept in 16
// steps of v_wmma_f32_16x16x4_f32 with register-resident accumulators.
__device__ __forceinline__ void mm64(float* __restrict__ C,
                                     const float* __restrict__ A,
                                     const float* __restrict__ B,
                                     bool transB) {
  const int lane = threadIdx.x & 31;
  const int wave = threadIdx.x >> 5;
  const int half = lane >> 4;       // which half-wave (K-pair select)
  const int l16  = lane & 15;
#pragma unroll
  for (int t = 0; t < 2; ++t) {
    const int tile = wave + t * 8;
    const int rb = (tile >> 2) << 4;   // tile row base
    const int cb = (tile & 3) << 4;    // tile col base
    v8f acc = {};
#pragma unroll
    for (int k0 = 0; k0 < 64; k0 += 4) {
      const int ka = k0 + 2 * half;    // even -> 8B-aligned v2f LDS loads
      v2f a = *(const v2f*)(A + (rb + l16) * SLDS + ka);   // A[m][ka..ka+1]
      v2f b;
      if (!transB) {
        b.x = B[ka * SLDS + cb + l16];                     // B[ka][n]
        b.y = B[(ka + 1) * SLDS + cb + l16];               // B[ka+1][n]
      } else {
        b = *(const v2f*)(B + (cb + l16) * SLDS + ka);     // B^T[ka][n]=B[n][ka]
      }
      acc = __builtin_amdgcn_wmma_f32_16x16x4_f32(
          /*neg_a=*/false, a, /*neg_b=*/false, b,
          /*c_mod=*/(short)0, acc, /*reuse_a=*/false, /*reuse_b=*/false);
    }
#pragma unroll
    for (int r = 0; r < 8; ++r)        // C VGPR r -> row r (+8 for half 1)
      C[(rb + r + 8 * half) * SLDS + cb + l16] = acc[r];
  }
}

// ---- block-wide Frobenius norm of an LDS matrix ---------------------------
__device__ __forceinline__ float block_frob(const float* M, float* red) {
  float s = 0.f;
  for (int e = threadIdx.x; e < 4096; e += NTHREADS) {
    float v = M[IDX(e)];
    s += v * v;
  }
  red[threadIdx.x] = s;
  __syncthreads();
  for (int off = NTHREADS >> 1; off > 0; off >>= 1) {
    if ((int)threadIdx.x < off) red[threadIdx.x] += red[threadIdx.x + off];
    __syncthreads();
  }
  float r = sqrtf(red[0]);
  __syncthreads();
  return r;
}

// ---- coupled Newton-Schulz: Y->sqrt(Y0), Z->Y0^{-1/2}; Y0 prescaled -------
__device__ void ns_iterate(float*& Y, float*& Z, float*& P,
                           float*& T1, float*& T2, int iters) {
  for (int it = 0; it < iters; ++it) {
    mm64(P, Z, Y, false);                       // P = Z @ Y
    __syncthreads();
    for (int e = threadIdx.x; e < 4096; e += NTHREADS) {
      int r = e >> 6, c = e & 63;
      float v = -0.5f * P[IDX(e)];
      if (r == c) v += 1.5f;
      P[IDX(e)] = v;                            // W = (3I - P)/2
    }
    __syncthreads();
    mm64(T1, Y, P, false);                      // Y' = Y @ W
    mm64(T2, P, Z, false);                      // Z' = W @ Z
    __syncthreads();
    float* t = Y; Y = T1; T1 = t;
    t = Z; Z = T2; T2 = t;
  }
}

// ---- in-place SPD principal square root (M <- sqrt(M)) --------------------
__device__ void spd_sqrt(float*& M, float*& Z, float*& P,
                         float*& T1, float*& T2, float* red, int iters) {
  float s = fmaxf(block_frob(M, red) * 0.5f, 1e-20f);  // lambda_max/s <= 2 < 3
  float invs = 1.0f / s;
  for (int e = threadIdx.x; e < 4096; e += NTHREADS) {
    int r = e >> 6, c = e & 63;
    M[IDX(e)] *= invs;
    Z[IDX(e)] = (r == c) ? 1.f : 0.f;
  }
  __syncthreads();
  ns_iterate(M, Z, P, T1, T2, iters);
  float rs = sqrtf(s);
  for (int e = threadIdx.x; e < 4096; e += NTHREADS)
    M[IDX(e)] *= rs;
  __syncthreads();
}

// ===========================================================================
// Kernel 1 (one block): B-derived constants -> ws[0..4095]=B^{1/2},
// ws[4096..8191]=B^{-1/2}, ws[8192..12287]=TT
// ===========================================================================
__global__ void __launch_bounds__(NTHREADS)
bary_consts_kernel(const float* __restrict__ Bm, float* __restrict__ ws) {
  __shared__ float s0[MAT_LDS], s1[MAT_LDS], s2[MAT_LDS],
                   s3[MAT_LDS], s4[MAT_LDS], s5[MAT_LDS];
  __shared__ float red[NTHREADS];
  float *Y = s0, *Z = s1, *P = s2, *T1 = s3, *T2 = s4, *L = s5;

  for (int e = threadIdx.x; e < 4096; e += NTHREADS)
    Y[IDX(e)] = Bm[e];
  __syncthreads();

  float s = fmaxf(block_frob(Y, red) * 0.5f, 1e-20f);
  float invs = 1.0f / s;
  for (int e = threadIdx.x; e < 4096; e += NTHREADS) {
    int r = e >> 6, c = e & 63;
    Y[IDX(e)] *= invs;
    Z[IDX(e)] = (r == c) ? 1.f : 0.f;
  }
  __syncthreads();
  ns_iterate(Y, Z, P, T1, T2, 22);          // Y=sqrt(B/s), Z=(B/s)^{-1/2}

  float rs = sqrtf(s), irs = 1.0f / rs;
  for (int e = threadIdx.x; e < 4096; e += NTHREADS) {
    ws[e] = Y[IDX(e)] * rs;                 // B^{1/2}
    float z = Z[IDX(e)] * irs;              // B^{-1/2}
    ws[4096 + e] = z;
    T1[IDX(e)] = z;
  }
  __syncthreads();
  mm64(L, T1, T1, false);                   // B^{-1} = B^{-1/2} @ B^{-1/2}
  __syncthreads();

  // In-LDS Cholesky of B^{-1}: L (lower) with B^{-1} = L L^T
  for (int j = 0; j < 64; ++j) {
    if (threadIdx.x == 0)
      L[j * SLDS + j] = sqrtf(fmaxf(L[j * SLDS + j], 1e-20f));
    __syncthreads();
    float d = L[j * SLDS + j];
    int i = (int)threadIdx.x;
    if (i > j && i < 64) L[i * SLDS + j] /= d;
    __syncthreads();
    if (i > j && i < 64) {
      float lij = L[i * SLDS + j];
      for (int c = j + 1; c <= i; ++c)
        L[i * SLDS + c] -= lij * L[c * SLDS + j];
    }
    __syncthreads();
  }

  // T = chol(B^{-1})^T (upper);  TT = T @ T
  for (int e = threadIdx.x; e < 4096; e += NTHREADS) {
    int r = e >> 6, c = e & 63;
    P[IDX(e)] = (r <= c) ? L[c * SLDS + r] : 0.f;
  }
  __syncthreads();
  mm64(T2, P, P, false);
  __syncthreads();
  for (int e = threadIdx.x; e < 4096; e += NTHREADS)
    ws[8192 + e] = T2[IDX(e)];
}

// ===========================================================================
// Kernel 2 (one block per sample): full per-sample pipeline.
// ===========================================================================
__global__ void __launch_bounds__(NTHREADS)
bary_norm_kernel(const float* __restrict__ X, const float* __restrict__ ws,
                 float* __restrict__ out) {
  __shared__ float b0[MAT_LDS], b1[MAT_LDS], b2[MAT_LDS],
                   b3[MAT_LDS], b4[MAT_LDS];
  __shared__ float red[NTHREADS];

  const float* Xi = X + (size_t)blockIdx.x * 4096;
  __builtin_prefetch(Xi, 0, 1);             // global_prefetch_b8

  float *M = b1, *Z = b0, *P = b2, *T1 = b3, *T2 = b4;

  // ---- stage A: M = B^{-1/2} X B^{-1/2} ----------------------------------
  for (int e = threadIdx.x; e < 4096; e += NTHREADS) {
    Z[IDX(e)] = ws[4096 + e];               // B^{-1/2}
    M[IDX(e)] = Xi[e];                      // X
  }
  __syncthreads();
  mm64(P, Z, M, false);                     // B^{-1/2} @ X
  __syncthreads();
  mm64(M, P, Z, false);                     // M
  __syncthreads();

  // ---- stage B: logm(M) via inverse scaling-and-squaring ------------------
  spd_sqrt(M, Z, P, T1, T2, red, 18);       // eigs -> near 1
  spd_sqrt(M, Z, P, T1, T2, red, 12);
  spd_sqrt(M, Z, P, T1, T2, red, 10);       // M = M0^(1/8)

  for (int e = threadIdx.x; e < 4096; e += NTHREADS) {
    int r = e >> 6, c = e & 63;
    if (r == c) M[IDX(e)] -= 1.f;           // E = M - I
  }
  __syncthreads();
  // H = Horner of p(E), log(I+E) = E * p(E), c_n = (-1)^n/(n+1), n=0..7
  for (int e = threadIdx.x; e < 4096; e += NTHREADS) {
    int r = e >> 6, c = e & 63;
    Z[IDX(e)] = (r == c) ? -0.125f : 0.f;   // c7 * I
  }
  __syncthreads();
  for (int n = 6; n >= 0; --n) {
    mm64(P, M, Z, false);                   // T = E @ H
    __syncthreads();
    float cn = ((n & 1) ? -1.f : 1.f) / (float)(n + 1);
    for (int e = threadIdx.x; e < 4096; e += NTHREADS) {
      int r = e >> 6, c = e & 63;
      Z[IDX(e)] = P[IDX(e)] + ((r == c) ? cn : 0.f);
    }
    __syncthreads();
  }
  mm64(T1, M, Z, false);                    // E @ p(E)
  __syncthreads();

  // ---- stage C: S = B^{1/2} (8*logM) B^{1/2} ------------------------------
  for (int e = threadIdx.x; e < 4096; e += NTHREADS) {
    T1[IDX(e)] *= 8.f;                      // logM = 2^3 * log(M^(1/8))
    T2[IDX(e)] = ws[e];                     // B^{1/2}
  }
  __syncthreads();
  mm64(P, T2, T1, false);
  __syncthreads();
  mm64(M, P, T2, false);                    // S
  for (int e = threadIdx.x; e < 4096; e += NTHREADS)
    Z[IDX(e)] = ws[8192 + e];               // TT
  __syncthreads();

  // ---- stage D: S_t = TT @ S @ TT^T ---------------------------------------
  mm64(P, Z, M, false);
  __syncthreads();
  mm64(T1, P, Z, true);                     // uses B-transposed frag loads
  __syncthreads();

  // ---- stage E: expm(S_t) by scaling-and-squaring -------------------------
  float* A = T1;
  float f = block_frob(A, red);
  int j = 0;
  float ff = f;
  while (ff > 0.5f && j < 12) { ff *= 0.5f; ++j; }
  float sc = exp2f((float)(-j));
  for (int e = threadIdx.x; e < 4096; e += NTHREADS)
    A[IDX(e)] *= sc;
  for (int e = threadIdx.x; e < 4096; e += NTHREADS) {
    int r = e >> 6, c = e & 63;
    Z[IDX(e)] = (r == c) ? 1.f : 0.f;       // H = I
  }
  __syncthreads();
  for (int k = 10; k >= 1; --k) {           // H = I + (A@H)/k  (Horner)
    mm64(P, A, Z, false);
    __syncthreads();
    float inv = 1.0f / (float)k;
    for (int e = threadIdx.x; e < 4096; e += NTHREADS) {
      int r = e >> 6, c = e & 63;
      Z[IDX(e)] = P[IDX(e)] * inv + ((r == c) ? 1.f : 0.f);
    }
    __syncthreads();
  }
  for (int q = 0; q < j; ++q) {             // repeated squaring
    mm64(P, Z, Z, false);
    __syncthreads();
    float* t = Z; Z = P; P = t;
  }

  float* O = out + (size_t)blockIdx.x * 4096;
  for (int e = threadIdx.x; e < 4096; e += NTHREADS)
    O[e] = Z[IDX(e)];
}

// ===========================================================================
extern "C" void kernel_launch(void* const* d_in, const int* in_sizes, int n_in,
                              void* d_out, int out_size, void* d_ws,
                              size_t ws_size, hipStream_t stream) {
  (void)n_in; (void)out_size; (void)ws_size;
  const float* X  = (const float*)d_in[0];   // (8192, 64, 64) f32
  const float* Bm = (const float*)d_in[1];   // (1, 64, 64) f32
  float* out = (float*)d_out;
  float* ws  = (float*)d_ws;                 // 3 * 4096 floats of constants
  const int batch = in_sizes[0] / 4096;

  bary_consts_kernel<<<1, NTHREADS, 0, stream>>>(Bm, ws);
  bary_norm_kernel<<<batch, NTHREADS, 0, stream>>>(X, ws, out);
}